// HashGridPrefixCommon_8796093022897
// MI455X (gfx1250) — compile-verified
//
#include <hip/hip_runtime.h>
#include <hip/hip_bf16.h>

typedef __attribute__((ext_vector_type(2))) float v2f;
typedef __attribute__((ext_vector_type(8))) float v8f;

// floor(16 * (1024/16)^(l/15)) for l = 0..15
__constant__ float kRES[16] = {16.f, 21.f, 27.f, 36.f, 48.f, 64.f, 84.f, 111.f,
                               147.f, 194.f, 256.f, 337.f, 445.f, 588.f, 776.f, 1024.f};

// ---------------------------------------------------------------------------
// Hashgrid encode: one thread per (point, level). 8 gathers from the 2MB table
// (L2-resident), trilinear blend, coalesced float2 write into the concat slot.
// dest layout: out[out_base + p*stride + l*2 .. +1]
// ---------------------------------------------------------------------------
__global__ void hashgrid_kernel(const float* __restrict__ pos,
                                const float* __restrict__ table,
                                float* __restrict__ out,
                                long long out_base, int stride, int npoints) {
  int tid = blockIdx.x * blockDim.x + threadIdx.x;
  int p = tid >> 4;
  int l = tid & 15;
  if (p >= npoints) return;

  float px = pos[p * 3 + 0], py = pos[p * 3 + 1], pz = pos[p * 3 + 2];
  float xn = fminf(fmaxf((px + 1.f) * 0.5f, 0.f), 1.f);
  float yn = fminf(fmaxf((py + 1.f) * 0.5f, 0.f), 1.f);
  float zn = fminf(fmaxf((pz + 1.f) * 0.5f, 0.f), 1.f);
  float res = kRES[l];
  float sx = xn * res, sy = yn * res, sz = zn * res;
  float fx = floorf(sx), fy = floorf(sy), fz = floorf(sz);
  float wx = sx - fx, wy = sy - fy, wz = sz - fz;
  unsigned cx = (unsigned)fx, cy = (unsigned)fy, cz = (unsigned)fz;

  const float* tl = table + (((size_t)l) << 14) * 2;  // level slice: 16384*2 floats
  float a0 = 0.f, a1 = 0.f;
#pragma unroll
  for (int c = 0; c < 8; ++c) {
    unsigned ox = (c >> 2) & 1u, oy = (c >> 1) & 1u, oz = c & 1u;
    unsigned h = (cx + ox) * 1u ^ (cy + oy) * 2654435761u ^ (cz + oz) * 805459861u;
    unsigned idx = h & 16383u;
    const float2 f = *(const float2*)(tl + (size_t)idx * 2);
    float wc = (ox ? wx : 1.f - wx) * (oy ? wy : 1.f - wy) * (oz ? wz : 1.f - wz);
    a0 += f.x * wc;
    a1 += f.y * wc;
  }
  float* dst = out + out_base + (long long)p * stride + l * 2;
  dst[0] = a0;
  dst[1] = a1;
}

// ---------------------------------------------------------------------------
// Lidar conv stack, one sample per wave32, fp32 WMMA 16x16x4.
//   x(128,8) -> conv(3,s2,8->16)+lrelu -> conv(3,s2,16->16)+lrelu
//            -> conv(3,s2,16->16) -> layernorm(256) -> lrelu
// SAME padding with stride 2 => pad (0,1): out[t] = sum_k in[2t+k], in[N]=0.
// im2col GEMMs: (64x24)(24x16), (32x48)(48x16), (16x48)(48x16).
// LDS rows padded (9 / 17 floats) to break stride-16/32 bank conflicts.
// A-frag (16x4 f32): lane<16 holds K={k0,k0+1}, lane>=16 K={k0+2,k0+3} of row M=lane%16.
// C/D (16x16 f32): acc[r] = row (r + 8*(lane>=16)), col lane%16.
// ---------------------------------------------------------------------------
__global__ __launch_bounds__(128) void lidar_wmma_kernel(
    const float* __restrict__ lidar, const float* __restrict__ w1,
    const float* __restrict__ b1, const float* __restrict__ w2,
    const float* __restrict__ b2, const float* __restrict__ w3,
    const float* __restrict__ b3, const float* __restrict__ g,
    const float* __restrict__ be, float* __restrict__ out, int col_base, int nsamp) {
  constexpr int XB = 129 * 9;   // x (rows 0..128, row 128 = zero pad), 9-float rows
  constexpr int O1N = 65 * 17;  // out1 (rows 0..64, row 64 = zero pad), 17-float rows
  constexpr int O2N = 33 * 17;  // out2 (rows 0..32, row 32 = zero pad)
  constexpr int PER = XB + O1N + O2N;
  __shared__ float lds[4 * PER];

  const int lane = threadIdx.x & 31;
  const int wv = threadIdx.x >> 5;
  const int half = lane >> 4;  // 0: lanes 0-15, 1: lanes 16-31
  const int lm = lane & 15;
  const int b = blockIdx.x * 4 + wv;
  const bool valid = (b < nsamp);

  float* xbuf = lds + wv * PER;
  float* o1 = xbuf + XB;
  float* o2 = o1 + O1N;

  // Load sample: lidar(b,4,128,2) -> x[t, a*2+c], coalesced 128B per iteration.
  if (valid) {
    const float* in = lidar + (size_t)b * 1024;
#pragma unroll
    for (int i = 0; i < 32; ++i) {
      int f = i * 32 + lane;  // f = a*256 + t*2 + c
      float v = in[f];
      xbuf[((f >> 1) & 127) * 9 + (f >> 8) * 2 + (f & 1)] = v;
    }
  }
  if (lane < 9) xbuf[128 * 9 + lane] = 0.f;
  if (lane < 17) { o1[64 * 17 + lane] = 0.f; o2[32 * 17 + lane] = 0.f; }
  __syncthreads();

  // ---- layer 1: (64x24)x(24x16), 4 M-tiles x 6 K-steps ----
  {
    float bb = b1[lm];
#pragma unroll
    for (int mt = 0; mt < 4; ++mt) {
      v8f acc = {0.f, 0.f, 0.f, 0.f, 0.f, 0.f, 0.f, 0.f};
      const int row = mt * 16 + lm;
#pragma unroll
      for (int ks = 0; ks < 6; ++ks) {
        const int k0 = ks * 4 + half * 2;
        v2f a, bf;
        a.x = xbuf[(2 * row + (k0 >> 3)) * 9 + (k0 & 7)];
        a.y = xbuf[(2 * row + ((k0 + 1) >> 3)) * 9 + ((k0 + 1) & 7)];
        bf.x = w1[k0 * 16 + lm];
        bf.y = w1[(k0 + 1) * 16 + lm];
        acc = __builtin_amdgcn_wmma_f32_16x16x4_f32(false, a, false, bf, (short)0, acc,
                                                    false, false);
      }
#pragma unroll
      for (int r = 0; r < 8; ++r) {
        float v = acc[r] + bb;
        v = v >= 0.f ? v : 0.01f * v;
        o1[(mt * 16 + r + half * 8) * 17 + lm] = v;
      }
    }
  }
  __syncthreads();

  // ---- layer 2: (32x48)x(48x16), 2 M-tiles x 12 K-steps ----
  {
    float bb = b2[lm];
#pragma unroll
    for (int mt = 0; mt < 2; ++mt) {
      v8f acc = {0.f, 0.f, 0.f, 0.f, 0.f, 0.f, 0.f, 0.f};
      const int row = mt * 16 + lm;
#pragma unroll
      for (int ks = 0; ks < 12; ++ks) {
        const int k0 = ks * 4 + half * 2;
        v2f a, bf;
        a.x = o1[(2 * row + (k0 >> 4)) * 17 + (k0 & 15)];
        a.y = o1[(2 * row + ((k0 + 1) >> 4)) * 17 + ((k0 + 1) & 15)];
        bf.x = w2[k0 * 16 + lm];
        bf.y = w2[(k0 + 1) * 16 + lm];
        acc = __builtin_amdgcn_wmma_f32_16x16x4_f32(false, a, false, bf, (short)0, acc,
                                                    false, false);
      }
#pragma unroll
      for (int r = 0; r < 8; ++r) {
        float v = acc[r] + bb;
        v = v >= 0.f ? v : 0.01f * v;
        o2[(mt * 16 + r + half * 8) * 17 + lm] = v;
      }
    }
  }
  __syncthreads();

  // ---- layer 3: (16x48)x(48x16), D stays in registers; fused layernorm ----
  {
    v8f acc = {0.f, 0.f, 0.f, 0.f, 0.f, 0.f, 0.f, 0.f};
#pragma unroll
    for (int ks = 0; ks < 12; ++ks) {
      const int k0 = ks * 4 + half * 2;
      v2f a, bf;
      a.x = o2[(2 * lm + (k0 >> 4)) * 17 + (k0 & 15)];
      a.y = o2[(2 * lm + ((k0 + 1) >> 4)) * 17 + ((k0 + 1) & 15)];
      bf.x = w3[k0 * 16 + lm];
      bf.y = w3[(k0 + 1) * 16 + lm];
      acc = __builtin_amdgcn_wmma_f32_16x16x4_f32(false, a, false, bf, (short)0, acc,
                                                  false, false);
    }
    float bb = b3[lm];
    float t[8];
    float s = 0.f, q = 0.f;
#pragma unroll
    for (int r = 0; r < 8; ++r) {
      t[r] = acc[r] + bb;
      s += t[r];
      q += t[r] * t[r];
    }
#pragma unroll
    for (int off = 16; off >= 1; off >>= 1) {
      s += __shfl_xor(s, off, 32);
      q += __shfl_xor(q, off, 32);
    }
    float mean = s * (1.f / 256.f);
    float var = q * (1.f / 256.f) - mean * mean;
    float rinv = rsqrtf(var + 1e-6f);
    if (valid) {
      float* dst = out + (size_t)b * 608 + col_base;
#pragma unroll
      for (int r = 0; r < 8; ++r) {
        int idx = (r + half * 8) * 16 + lm;  // flat = t3*16 + cout
        float v = (t[r] - mean) * rinv * g[idx] + be[idx];
        v = v >= 0.f ? v : 0.01f * v;
        dst[idx] = v;
      }
    }
  }
}

// ---------------------------------------------------------------------------
// Strided float4 row copy: src rows contiguous, dst rows at dst_off + r*stride.
// ---------------------------------------------------------------------------
__global__ void copy_rows4(const float* __restrict__ src, float* __restrict__ dst,
                           long long nrows, int row_len4, long long dst_stride,
                           long long dst_off) {
  long long t = (long long)blockIdx.x * blockDim.x + threadIdx.x;
  if (t >= nrows * (long long)row_len4) return;
  long long r = t / row_len4;
  int c = (int)(t % row_len4);
  const float4 v = ((const float4*)src)[r * (long long)row_len4 + c];
  *((float4*)(dst + dst_off + r * dst_stride) + c) = v;
}

extern "C" void kernel_launch(void* const* d_in, const int* in_sizes, int n_in,
                              void* d_out, int out_size, void* d_ws, size_t ws_size,
                              hipStream_t stream) {
  (void)n_in; (void)out_size; (void)d_ws; (void)ws_size;
  const float* self_pos   = (const float*)d_in[0];
  const float* tm_pos     = (const float*)d_in[1];
  const float* op_pos     = (const float*)d_in[2];
  const float* ol_pos     = (const float*)d_in[3];
  const float* self_vec   = (const float*)d_in[4];
  const float* fwd_lidar  = (const float*)d_in[5];
  const float* rear_lidar = (const float*)d_in[6];
  const float* teammates  = (const float*)d_in[7];
  const float* opponents  = (const float*)d_in[8];
  const float* opp_last   = (const float*)d_in[9];
  const float* masks      = (const float*)d_in[10];
  const float* amap       = (const float*)d_in[11];
  const float* umap       = (const float*)d_in[12];
  const float* table      = (const float*)d_in[13];
  const float* fw1 = (const float*)d_in[14];
  const float* fb1 = (const float*)d_in[15];
  const float* fw2 = (const float*)d_in[16];
  const float* fb2 = (const float*)d_in[17];
  const float* fw3 = (const float*)d_in[18];
  const float* fb3 = (const float*)d_in[19];
  const float* fg  = (const float*)d_in[20];
  const float* fbe = (const float*)d_in[21];
  const float* rw1 = (const float*)d_in[22];
  const float* rb1 = (const float*)d_in[23];
  const float* rw2 = (const float*)d_in[24];
  const float* rb2 = (const float*)d_in[25];
  const float* rw3 = (const float*)d_in[26];
  const float* rb3 = (const float*)d_in[27];
  const float* rg  = (const float*)d_in[28];
  const float* rbe = (const float*)d_in[29];
  float* out = (float*)d_out;

  const long long B = in_sizes[0] / 3;
  // Output tuple layout (flat floats):
  const long long tm_base   = B * 608;
  const long long op_base   = tm_base + B * 5 * 64;
  const long long ol_base   = op_base + B * 6 * 64;
  const long long mask_base = ol_base + B * 6 * 64;
  const long long amap_base = mask_base + B * 6;
  const long long umap_base = amap_base + B * 16 * 16 * 4;

  auto cdiv = [](long long a, long long d) { return (unsigned)((a + d - 1) / d); };

  // Hashgrid encodings written straight into their concat slots.
  hashgrid_kernel<<<cdiv(B * 16, 256), 256, 0, stream>>>(self_pos, table, out, 0, 608, (int)B);
  hashgrid_kernel<<<cdiv(B * 5 * 16, 256), 256, 0, stream>>>(tm_pos, table, out, tm_base, 64, (int)(B * 5));
  hashgrid_kernel<<<cdiv(B * 6 * 16, 256), 256, 0, stream>>>(op_pos, table, out, op_base, 64, (int)(B * 6));
  hashgrid_kernel<<<cdiv(B * 6 * 16, 256), 256, 0, stream>>>(ol_pos, table, out, ol_base, 64, (int)(B * 6));

  // Lidar conv stacks (fp32 WMMA), fused into self_ob columns 96:352 / 352:608.
  lidar_wmma_kernel<<<cdiv(B, 4), 128, 0, stream>>>(fwd_lidar, fw1, fb1, fw2, fb2, fw3, fb3,
                                                    fg, fbe, out, 96, (int)B);
  lidar_wmma_kernel<<<cdiv(B, 4), 128, 0, stream>>>(rear_lidar, rw1, rb1, rw2, rb2, rw3, rb3,
                                                    rg, rbe, out, 352, (int)B);

  // Concat copies / passthroughs (float4 vectorized).
  copy_rows4<<<cdiv(B * 16, 256), 256, 0, stream>>>(self_vec, out, B, 16, 608, 32);
  copy_rows4<<<cdiv(B * 5 * 8, 256), 256, 0, stream>>>(teammates, out, B * 5, 8, 64, tm_base + 32);
  copy_rows4<<<cdiv(B * 6 * 8, 256), 256, 0, stream>>>(opponents, out, B * 6, 8, 64, op_base + 32);
  copy_rows4<<<cdiv(B * 6 * 8, 256), 256, 0, stream>>>(opp_last, out, B * 6, 8, 64, ol_base + 32);
  copy_rows4<<<cdiv(B * 6 / 4, 256), 256, 0, stream>>>(masks, out, 1, (int)(B * 6 / 4), 0, mask_base);
  copy_rows4<<<cdiv(B * 256, 256), 256, 0, stream>>>(amap, out, 1, (int)(B * 256), 0, amap_base);
  copy_rows4<<<cdiv(B * 256, 256), 256, 0, stream>>>(umap, out, 1, (int)(B * 256), 0, umap_base);
}